// ChamferLoss_27642409517713
// MI455X (gfx1250) — compile-verified
//
#include <hip/hip_runtime.h>
#include <hip/hip_bf16.h>
#include <math.h>

typedef __attribute__((ext_vector_type(2))) float v2f;
typedef __attribute__((ext_vector_type(8))) float v8f;

#define BATCH 16
#define MPTS  4096
#define NPTS  4096
#define EPSF  1e-12f

// ---------------------------------------------------------------------------
// out[0] = 0
// ---------------------------------------------------------------------------
__global__ void zero_out_kernel(float* __restrict__ out) {
    if (threadIdx.x == 0 && blockIdx.x == 0) out[0] = 0.0f;
}

// ---------------------------------------------------------------------------
// Pack kp [B,3,npts] into WMMA-fragment-ready layouts:
//   aP[p] = [-2x, -2y, -2z, 1.0]   (A-matrix row: K0..K3)
//   bP[p] = [  x,   y,   z,  sq]   (B-matrix col: K0..K3, sq = ||p||^2)
//   sq[p] = ||p||^2                (epilogue)
// Lanes 0-15 of a wave read the (K0,K1) pair at byte offset 0, lanes 16-31
// the (K2,K3) pair at offset 8 -> one b64 load builds a fragment, no selects.
// ---------------------------------------------------------------------------
__global__ void pack_kernel(const float* __restrict__ kp,
                            float4* __restrict__ aP,
                            float4* __restrict__ bP,
                            float*  __restrict__ sq, int npts) {
    int idx = blockIdx.x * blockDim.x + threadIdx.x;   // [0, B*npts)
    int b = idx / npts;
    int p = idx - b * npts;
    const float* base = kp + (size_t)b * 3 * npts;
    float x = base[p];
    float y = base[npts + p];
    float z = base[2 * npts + p];
    float s = x * x + y * y + z * z;
    aP[idx] = make_float4(-2.0f * x, -2.0f * y, -2.0f * z, 1.0f);
    bP[idx] = make_float4(x, y, z, s);
    sq[idx] = s;
}

// ---------------------------------------------------------------------------
// One direction of the Chamfer loss.
//
// The WMMA does ALL the arithmetic for the 16x16 tile:
//   A (16x4):  row m -> [-2x_m, -2y_m, -2z_m, 1.0]
//   B (4x16):  col n -> [x_n, y_n, z_n, ||b_n||^2]^T
//   D = 0 + A*B = ||b_n||^2 - 2 a_m.b_n     (inline C=0)
// Per-row min is translation-invariant, so ||a_m||^2 is added once per row
// in the epilogue; sqrt and the EPS clamp also commute with min.
// The loop keeps a fused (min, sigma-at-argmin) pair -> no argmin+gather pass.
// Fragments come straight from memory as b64 loads (see pack_kernel); all
// lane-dependent addressing is loop-invariant, EXEC is all-ones throughout.
// ---------------------------------------------------------------------------
__global__ __launch_bounds__(256) void chamfer_dir_kernel(
    const float4* __restrict__ aPk,  // A-form pack of the "query" set
    const float4* __restrict__ bPk,  // B-form pack of the "target" set
    const float*  __restrict__ sqA,  // ||a||^2 of the query set
    const float*  __restrict__ sigA, const float* __restrict__ sigB,
    float* __restrict__ out, int Mrows, int Ncols, float scale)
{
    const int wave = threadIdx.x >> 5;
    const int lane = threadIdx.x & 31;
    const int lo   = lane & 15;        // column / point-in-tile index
    const int hi   = lane >> 4;        // half-wave: (K0,K1) vs (K2,K3) pair

    const int tilesPerBatch = Mrows >> 4;
    const int tile = blockIdx.x * 8 + wave;          // 8 waves per block
    const int b    = tile / tilesPerBatch;
    const int mt   = tile - b * tilesPerBatch;
    const int m0   = mt << 4;

    const v2f*   Afrag = (const v2f*)(aPk + (size_t)b * Mrows);
    const v2f*   Bfrag = (const v2f*)(bPk + (size_t)b * Ncols);
    const float* sq1   = sqA  + (size_t)b * Mrows;
    const float* s1    = sigA + (size_t)b * Mrows;
    const float* s2    = sigB + (size_t)b * Ncols;

    // A fragment: one b64 load, already scaled/padded in memory
    v2f a = Afrag[(m0 + lo) * 2 + hi];

    float minv[8], msig[8];   // min of (||b||^2 - 2 a.b) + sigma at argmin
    #pragma unroll
    for (int k = 0; k < 8; ++k) { minv[k] = 3.4e38f; msig[k] = 1.0f; }

    const int laneSlot = lo * 2 + hi;  // loop-invariant fragment address part

    for (int n0 = 0; n0 < Ncols; n0 += 16) {
        // speculative prefetch of the next B tile (global_prefetch_b8)
        __builtin_prefetch(&Bfrag[(n0 + 16) * 2 + laneSlot], 0, 3);

        v2f   bf   = Bfrag[n0 * 2 + laneSlot];   // fragment pair, one b64
        float sg2s = s2[n0 + lo];                // sigma of this lane's column

        v8f c = {};                              // inline zero accumulator
        c = __builtin_amdgcn_wmma_f32_16x16x4_f32(
                /*neg_a=*/false, a, /*neg_b=*/false, bf,
                /*c_mod=*/(short)0, c, /*reuse_a=*/false, /*reuse_b=*/false);

        #pragma unroll
        for (int k = 0; k < 8; ++k) {
            float v = c[k];
            if (v < minv[k]) { minv[k] = v; msig[k] = sg2s; }
        }
    }

    // Row-min reduction across the 16 lanes of each half-wave, carrying sigma.
    #pragma unroll
    for (int mask = 1; mask < 16; mask <<= 1) {
        #pragma unroll
        for (int k = 0; k < 8; ++k) {
            float ov = __shfl_xor(minv[k], mask, 32);
            float os = __shfl_xor(msig[k], mask, 32);
            if (ov < minv[k]) { minv[k] = ov; msig[k] = os; }
        }
    }

    // Lane 0 of each half-wave owns 8 rows: restore ||a||^2, sqrt, finish.
    if (lo == 0) {
        float acc = 0.0f;
        #pragma unroll
        for (int k = 0; k < 8; ++k) {
            int m = m0 + k + 8 * hi;
            float d2 = minv[k] + sq1[m];                // + ||a_m||^2
            float d  = __builtin_sqrtf(fmaxf(d2, EPSF));
            float sg = (s1[m] + msig[k]) * 0.5f;
            acc += logf(sg) + d / sg;
        }
        atomicAdd(out, acc * scale);
    }
}

// ---------------------------------------------------------------------------
// kernel_launch
// inputs: keypoints1 [B,3,M] f32, keypoints2 [B,3,N] f32,
//         sigma1 [B,M] f32, sigma2 [B,N] f32     output: 1 float
// workspace: per set: A-pack (1 MB) + B-pack (1 MB) + sq (256 KB)  => 4.5 MB
// ---------------------------------------------------------------------------
extern "C" void kernel_launch(void* const* d_in, const int* in_sizes, int n_in,
                              void* d_out, int out_size, void* d_ws, size_t ws_size,
                              hipStream_t stream) {
    const float* k1  = (const float*)d_in[0];
    const float* k2  = (const float*)d_in[1];
    const float* sg1 = (const float*)d_in[2];
    const float* sg2 = (const float*)d_in[3];
    float* out = (float*)d_out;

    const size_t P = (size_t)BATCH * MPTS;   // == BATCH * NPTS
    float4* aP1 = (float4*)d_ws;             // A-form of set 1
    float4* bP1 = aP1 + P;                   // B-form of set 1
    float4* aP2 = bP1 + P;                   // A-form of set 2
    float4* bP2 = aP2 + P;                   // B-form of set 2
    float*  sq1 = (float*)(bP2 + P);         // ||.||^2 of set 1
    float*  sq2 = sq1 + P;                   // ||.||^2 of set 2

    zero_out_kernel<<<1, 1, 0, stream>>>(out);

    pack_kernel<<<(BATCH * MPTS) / 256, 256, 0, stream>>>(k1, aP1, bP1, sq1, MPTS);
    pack_kernel<<<(BATCH * NPTS) / 256, 256, 0, stream>>>(k2, aP2, bP2, sq2, NPTS);

    const float scaleF = 1.0f / (float)(BATCH * MPTS);
    const float scaleB = 1.0f / (float)(BATCH * NPTS);

    // forward: for each point in set1, nearest in set2
    chamfer_dir_kernel<<<(BATCH * (MPTS / 16)) / 8, 256, 0, stream>>>(
        aP1, bP2, sq1, sg1, sg2, out, MPTS, NPTS, scaleF);

    // backward: roles swapped
    chamfer_dir_kernel<<<(BATCH * (NPTS / 16)) / 8, 256, 0, stream>>>(
        aP2, bP1, sq2, sg2, sg1, out, NPTS, MPTS, scaleB);
}